// PointPillarFeatureNet_22273700397341
// MI455X (gfx1250) — compile-verified
//
#include <hip/hip_runtime.h>
#include <hip/hip_bf16.h>
#include <cstdint>

#define PP 30000
#define NPT 100
#define COUT 64
#define EPSB 0.001f

typedef __attribute__((ext_vector_type(2))) float v2f;
typedef __attribute__((ext_vector_type(8))) float v8f;

__device__ __forceinline__ int tri9(int i, int j) { return i*9 - (i*(i-1))/2 + (j-i); } // i<=j, 45 entries
__device__ __forceinline__ int tri4(int i, int j) { return i*4 - (i*(i-1))/2 + (j-i); } // i<=j, 10 entries

// ---------------- Kernel A: global BN moment accumulation (9-vector + 9x9 sym) ----
__global__ void __launch_bounds__(256)
pfn_stats_kernel(const float* __restrict__ feat, const int* __restrict__ npts,
                 const int* __restrict__ coords, float* __restrict__ gstats) {
  __shared__ float sacc[54];
  const int tid = threadIdx.x;
  if (tid < 54) sacc[tid] = 0.f;
  __syncthreads();
  const int p = blockIdx.x * blockDim.x + tid;
  if (p < PP) {
    int V = npts[p]; V = (V > NPT) ? NPT : V; V = (V < 1) ? 1 : V;
    const float* fp = feat + (size_t)p * NPT * 4;
    float sg[4] = {0.f, 0.f, 0.f, 0.f};
    float gg[10];
    #pragma unroll
    for (int k = 0; k < 10; ++k) gg[k] = 0.f;
    for (int m = 0; m < V; ++m) {
      float g[4] = {fp[m*4+0], fp[m*4+1], fp[m*4+2], fp[m*4+3]};
      #pragma unroll
      for (int i = 0; i < 4; ++i) sg[i] += g[i];
      #pragma unroll
      for (int i = 0; i < 4; ++i)
        #pragma unroll
        for (int j = i; j < 4; ++j) gg[tri4(i, j)] += g[i] * g[j];
    }
    const float Vf = (float)V;
    const float mx = sg[0] / Vf, my = sg[1] / Vf, mz = sg[2] / Vf;
    const float Cx = (float)coords[p*4+3] * 0.16f + 0.08f;
    const float Cy = (float)coords[p*4+2] * 0.16f + (-39.6f);
    const float u[5] = {-mx, -my, -mz, -Cx, -Cy};
    // sum of h = [g(4), u(5)] over valid points (u constant per point)
    #pragma unroll
    for (int i = 0; i < 4; ++i) atomicAdd(&sacc[i], sg[i]);
    #pragma unroll
    for (int k = 0; k < 5; ++k) atomicAdd(&sacc[4+k], Vf * u[k]);
    // S = sum h h^T (upper triangle)
    #pragma unroll
    for (int i = 0; i < 9; ++i)
      #pragma unroll
      for (int j = i; j < 9; ++j) {
        float v;
        if (j < 4)      v = gg[tri4(i, j)];
        else if (i < 4) v = sg[i] * u[j-4];
        else            v = Vf * u[i-4] * u[j-4];
        atomicAdd(&sacc[9 + tri9(i, j)], v);
      }
  }
  __syncthreads();
  if (tid < 54) atomicAdd(&gstats[tid], sacc[tid]);
}

// ---------------- Kernel B: one wave per pillar; WMMA f32 16x16x4 GEMM + max/min ----
// Mask-free inner loop: rows m >= V (masked or pad) load row V-1 instead, producing an
// exact duplicate of x[V-1] in D — harmless for max/min. Masked rows' x==0 contribution
// is folded into the max/min initializers.
__global__ void __launch_bounds__(256)
pfn_gemm_minmax_kernel(const float* __restrict__ feat, const int* __restrict__ npts,
                       const int* __restrict__ coords, const float* __restrict__ W,
                       float* __restrict__ xmax, float* __restrict__ xmin) {
  const int lane = threadIdx.x & 31;
  const int wave = threadIdx.x >> 5;
  const int p = blockIdx.x * 8 + wave;
  if (p >= PP) return;                       // wave-uniform: EXEC stays full for WMMA
  int V = npts[p]; V = (V > NPT) ? NPT : V; V = (V < 1) ? 1 : V;
  const float Vf = (float)V;
  const float* fp = feat + (size_t)p * NPT * 4;

  // per-pillar mean of xyz over valid points (cooperative wave reduction)
  float sx = 0.f, sy = 0.f, sz = 0.f;
  #pragma unroll
  for (int j = 0; j < 4; ++j) {
    int m = lane + 32*j;
    if (m < V) { sx += fp[m*4+0]; sy += fp[m*4+1]; sz += fp[m*4+2]; }
  }
  #pragma unroll
  for (int off = 16; off >= 1; off >>= 1) {
    sx += __shfl_xor(sx, off, 32);
    sy += __shfl_xor(sy, off, 32);
    sz += __shfl_xor(sz, off, 32);
  }
  const float mx = sx / Vf, my = sy / Vf, mz = sz / Vf;
  const float Cx = (float)coords[p*4+3] * 0.16f + 0.08f;
  const float Cy = (float)coords[p*4+2] * 0.16f + (-39.6f);

  const int col = lane & 15;
  const bool hi = (lane >= 16);
  const float PINF = __builtin_inff();
  const float NINF = -__builtin_inff();

  // Folded B operand (Wp 4x64) + per-pillar per-channel constant K (goes into WMMA C).
  // B 4x16 layout mirrors documented 32-bit A 16x4: VGPR0 = K0|K2, VGPR1 = K1|K3 halves.
  v2f bmat[4];
  v8f cmat[4];
  float vmax[4], vmin[4];
  #pragma unroll
  for (int t = 0; t < 4; ++t) {
    const int c = t*16 + col;
    const float* w = W + c*9;
    const float wp0 = w[0] + w[4] + w[7];
    const float wp1 = w[1] + w[5] + w[8];
    const float wp2 = w[2] + w[6];
    const float wp3 = w[3];
    bmat[t].x = hi ? wp2 : wp0;
    bmat[t].y = hi ? wp3 : wp1;
    const float Kc = -(mx*w[4] + my*w[5] + mz*w[6] + Cx*w[7] + Cy*w[8]);
    #pragma unroll
    for (int j = 0; j < 8; ++j) cmat[t][j] = Kc;   // constant folded into C operand
    // masked rows (n >= V) contribute x == 0 to the reference max: fold into init
    vmax[t] = (V < NPT) ? 0.f : NINF;
    vmin[t] = (V < NPT) ? 0.f : PINF;
  }

  // 7 M-tiles of 16 points x 4 N-tiles of 16 channels, K=4 in a single WMMA each.
  for (int tile = 0; tile < 7; ++tile) {
    const int m = tile*16 + col;             // A: lanes 0-15 carry K0..1, 16-31 carry K2..3
    const int mc = (m < V) ? m : (V - 1);    // clamp: out-of-range rows duplicate row V-1
    const float* ap = fp + mc*4 + (hi ? 2 : 0);
    v2f a; a.x = ap[0]; a.y = ap[1];
    #pragma unroll
    for (int t = 0; t < 4; ++t) {
      v8f d = __builtin_amdgcn_wmma_f32_16x16x4_f32(
          false, a, false, bmat[t], (short)0, cmat[t], false, false);
      #pragma unroll
      for (int j = 0; j < 8; ++j) {          // mask-free: duplicates can't change max/min
        vmax[t] = fmaxf(vmax[t], d[j]);
        vmin[t] = fminf(vmin[t], d[j]);
      }
    }
  }

  #pragma unroll
  for (int t = 0; t < 4; ++t) {
    vmax[t] = fmaxf(vmax[t], __shfl_xor(vmax[t], 16, 32));
    vmin[t] = fminf(vmin[t], __shfl_xor(vmin[t], 16, 32));
    if (!hi) {
      xmax[(size_t)p*COUT + t*16 + col] = vmax[t];
      xmin[(size_t)p*COUT + t*16 + col] = vmin[t];
    }
  }
}

// ---------------- Kernel C: fold 54 moments into per-channel BN affine (a, b) ----
__global__ void pfn_bnparams_kernel(const float* __restrict__ gstats,
                                    const float* __restrict__ W,
                                    const float* __restrict__ gamma,
                                    const float* __restrict__ beta,
                                    float* __restrict__ ab) {
  const int c = threadIdx.x;
  if (c >= COUT) return;
  const float* w = W + c*9;
  float wm[9];
  wm[0] = w[0] + w[4] + w[7];
  wm[1] = w[1] + w[5] + w[8];
  wm[2] = w[2] + w[6];
  wm[3] = w[3];
  #pragma unroll
  for (int k = 4; k < 9; ++k) wm[k] = w[k];
  const float inv = 1.0f / ((float)PP * (float)NPT);
  float sx = 0.f;
  #pragma unroll
  for (int i = 0; i < 9; ++i) sx += gstats[i] * wm[i];
  const float mean = sx * inv;
  const float* S = gstats + 9;
  float e2 = 0.f;
  #pragma unroll
  for (int i = 0; i < 9; ++i)
    #pragma unroll
    for (int j = i; j < 9; ++j) {
      const float f = (i == j) ? 1.f : 2.f;
      e2 += f * S[tri9(i, j)] * wm[i] * wm[j];
    }
  e2 *= inv;
  const float var = e2 - mean * mean;
  const float a = gamma[c] * rsqrtf(var + EPSB);
  const float b = beta[c] - mean * a;
  ab[c] = a;
  ab[COUT + c] = b;
}

// ---------------- Kernel D: out = relu(a * (a>=0 ? xmax : xmin) + b) ----
__global__ void __launch_bounds__(256)
pfn_out_kernel(const float* __restrict__ ab, const float* __restrict__ xmax,
               const float* __restrict__ xmin, float* __restrict__ out) {
  const int idx = blockIdx.x * blockDim.x + threadIdx.x;
  if (idx >= PP * COUT) return;
  const int c = idx & (COUT - 1);
  const float a = ab[c];
  const float b = ab[COUT + c];
  const float v = (a >= 0.f) ? xmax[idx] : xmin[idx];
  out[idx] = fmaxf(fmaf(a, v, b), 0.f);
}

extern "C" void kernel_launch(void* const* d_in, const int* in_sizes, int n_in,
                              void* d_out, int out_size, void* d_ws, size_t ws_size,
                              hipStream_t stream) {
  const float* feat   = (const float*)d_in[0];
  const int*   npts   = (const int*)d_in[1];
  const int*   coords = (const int*)d_in[2];
  const float* W      = (const float*)d_in[3];
  const float* gamma  = (const float*)d_in[4];
  const float* beta   = (const float*)d_in[5];
  float* out = (float*)d_out;

  float* ws     = (float*)d_ws;
  float* gstats = ws;                                   // 54 floats
  float* ab     = ws + 64;                              // a[64] then b[64]
  float* xmax   = ws + 256;                             // P*64
  float* xmin   = ws + 256 + (size_t)PP * COUT;         // P*64

  hipMemsetAsync(d_ws, 0, 256 * sizeof(float), stream); // zero the accumulators each call

  pfn_stats_kernel<<<(PP + 255) / 256, 256, 0, stream>>>(feat, npts, coords, gstats);
  pfn_gemm_minmax_kernel<<<(PP + 7) / 8, 256, 0, stream>>>(feat, npts, coords, W, xmax, xmin);
  pfn_bnparams_kernel<<<1, 64, 0, stream>>>(gstats, W, gamma, beta, ab);
  pfn_out_kernel<<<(PP * COUT + 255) / 256, 256, 0, stream>>>(ab, xmax, xmin, out);
}